// BoundaryLoss_24068996726832
// MI455X (gfx1250) — compile-verified
//
#include <hip/hip_runtime.h>
#include <math.h>

// ---------- problem constants ----------
#define Bn   4
#define Cn   64
#define Hn   512
#define Wn   512
#define HWn  (Hn * Wn)          // 262144
#define TEMP 0.05f
#define EPSF 1e-8f

typedef __attribute__((ext_vector_type(2))) float v2f;
typedef __attribute__((ext_vector_type(4))) float v4f;
typedef __attribute__((ext_vector_type(8))) float v8f;

// ---------------------------------------------------------------------------
// Kernel 0: zero the small accumulator region (fs_bd[256], fs_bg[256], msums)
// ---------------------------------------------------------------------------
__global__ void bl_zero_kernel(float* acc, int n) {
    int i = blockIdx.x * blockDim.x + threadIdx.x;
    if (i < n) acc[i] = 0.0f;
}

// ---------------------------------------------------------------------------
// Kernel 1: boundary/background masks via a single 25-point L1-ball(2) stencil
// (dilate x2 / erode x2 with 3-D cross == dilate/erode with L1 ball radius 2,
//  padding False; the batch axis participates, matching the rank-3 scipy op)
// Also reduces mask sums per batch (each 256-thread block lies in one batch).
// ---------------------------------------------------------------------------
__device__ __forceinline__ bool bl_tget(const int* __restrict__ t, int b, int h, int w) {
    if ((unsigned)b >= Bn || (unsigned)h >= Hn || (unsigned)w >= Wn) return false;
    return t[((size_t)b * Hn + h) * Wn + w] != 0;
}

__global__ void bl_mask_kernel(const int* __restrict__ tgt,
                               float* __restrict__ bd, float* __restrict__ bg,
                               float* __restrict__ msum_bd, float* __restrict__ msum_bg) {
    const int tid = threadIdx.x;
    const int idx = blockIdx.x * 256 + tid;           // < Bn*HWn
    const int b   = idx / HWn;
    const int rem = idx - b * HWn;
    const int h   = rem >> 9;
    const int w   = rem & (Wn - 1);

    bool dil = false, ero = true;
#pragma unroll
    for (int db = -2; db <= 2; ++db) {
        const int rb = 2 - (db < 0 ? -db : db);
#pragma unroll
        for (int dh = -2; dh <= 2; ++dh) {
            if (dh < -rb || dh > rb) continue;
            const int rw = rb - (dh < 0 ? -dh : dh);
#pragma unroll
            for (int dw = -2; dw <= 2; ++dw) {
                if (dw < -rw || dw > rw) continue;
                bool v = bl_tget(tgt, b + db, h + dh, w + dw);
                dil |= v;
                ero &= v;
            }
        }
    }
    const bool boundary = dil ^ ero;                  // eroded subset of dilated
    const bool center   = tgt[idx] != 0;
    const float fb = boundary ? 1.0f : 0.0f;
    const float fg = (boundary && !center) ? 1.0f : 0.0f;
    bd[idx] = fb;
    bg[idx] = fg;

    // block reduction of the two mask sums (uniform b per block: HWn % 256 == 0)
    __shared__ float sA[256], sB[256];
    sA[tid] = fb; sB[tid] = fg;
    __syncthreads();
    for (int s = 128; s > 0; s >>= 1) {
        if (tid < s) { sA[tid] += sA[tid + s]; sB[tid] += sB[tid + s]; }
        __syncthreads();
    }
    if (tid == 0) {
        atomicAdd(&msum_bd[b], sA[0]);
        atomicAdd(&msum_bg[b], sB[0]);
    }
}

// ---------------------------------------------------------------------------
// Kernel 2: HBM-bound masked reduction fs[b,c] = sum_hw feat*mask, two masks.
// 256 MB of features streamed ONCE with non-temporal b128 loads (TH=NT keeps
// the 8 MB of masks resident in L2 across the 64 channel passes -> HBM traffic
// stays ~256 MB, ~11 us at 23.3 TB/s). grid = (B*C, 64 hw-splits).
// ---------------------------------------------------------------------------
__global__ __launch_bounds__(256)
void bl_reduce_kernel(const float* __restrict__ feat,
                      const float* __restrict__ bd, const float* __restrict__ bg,
                      float* __restrict__ fs_bd, float* __restrict__ fs_bg) {
    const int tid = threadIdx.x;
    const int bc  = blockIdx.x;                       // 0..255 : b*64 + c
    const int b   = bc >> 6;
    const int hw0 = blockIdx.y * (HWn / 64);          // 4096-float chunk

    const v4f* f4  = (const v4f*)(feat + (size_t)bc * HWn + hw0);
    const v4f* mb4 = (const v4f*)(bd   + (size_t)b  * HWn + hw0);
    const v4f* mg4 = (const v4f*)(bg   + (size_t)b  * HWn + hw0);

    float abd = 0.0f, abg = 0.0f;
#pragma unroll
    for (int i = 0; i < 4; ++i) {                     // 4 x 256 threads x float4 = 4096
        const int k = i * 256 + tid;
        if (i < 3) __builtin_prefetch(&f4[k + 256], 0, 0);   // global_prefetch_b8
        v4f f = __builtin_nontemporal_load(&f4[k]);          // TH=NT: rinse features
        v4f m = mb4[k];
        v4f g = mg4[k];
        abd += f.x * m.x + f.y * m.y + f.z * m.z + f.w * m.w;
        abg += f.x * g.x + f.y * g.y + f.z * g.z + f.w * g.w;
    }

    __shared__ float sA[256], sB[256];
    sA[tid] = abd; sB[tid] = abg;
    __syncthreads();
    for (int s = 128; s > 0; s >>= 1) {
        if (tid < s) { sA[tid] += sA[tid + s]; sB[tid] += sB[tid + s]; }
        __syncthreads();
    }
    if (tid == 0) {
        atomicAdd(&fs_bd[bc], sA[0]);
        atomicAdd(&fs_bg[bc], sB[0]);
    }
}

// ---------------------------------------------------------------------------
// Kernel 3: single wave32. Gram matrix G = R * R^T of the stacked rows
// R = [fs_bd(4x64); fs_bg(4x64); 0(8x64)] via 16 x V_WMMA_F32_16X16X4_F32
// (K = 64 in steps of 4). For f32 16x16x4: lane l<16 holds K={0,1}, lane>=16
// holds K={2,3} of row/col (l&15) -> the A fragment and the B(=A^T) fragment
// are the same registers. Then the 16-term scalar loss on lane 0.
// ---------------------------------------------------------------------------
__global__ void bl_finalize_kernel(const float* __restrict__ fs_bd,
                                   const float* __restrict__ fs_bg,
                                   const float* __restrict__ msum_bd,
                                   const float* __restrict__ msum_bg,
                                   float* __restrict__ out) {
    const int lane = threadIdx.x;         // 0..31, one full wave, EXEC all ones
    const int m    = lane & 15;
    const int hi   = lane >> 4;
    const int kk0  = hi * 2;

    const float* rowp = (m < 4) ? (fs_bd + m * Cn) : (fs_bg + ((m - 4) & 3) * Cn);
    const float keep  = (m < 8) ? 1.0f : 0.0f;

    v8f acc = {0.f, 0.f, 0.f, 0.f, 0.f, 0.f, 0.f, 0.f};
#pragma unroll
    for (int k = 0; k < Cn; k += 4) {
        v2f a;
        a.x = keep * rowp[k + kk0];
        a.y = keep * rowp[k + kk0 + 1];
        // D = A * A^T + D  : same fragment serves as A (16x4) and B (4x16)
        acc = __builtin_amdgcn_wmma_f32_16x16x4_f32(
            /*neg_a=*/false, a, /*neg_b=*/false, a,
            /*c_mod=*/(short)0, acc, /*reuse_a=*/false, /*reuse_b=*/false);
    }

    // scatter D (16x16 f32: vgpr v, lane<16 -> M=v,N=lane; lane>=16 -> M=v+8)
    __shared__ float G[16 * 16];
#pragma unroll
    for (int v = 0; v < 8; ++v) {
        const int M = hi ? (v + 8) : v;
        G[M * 16 + m] = acc[v];
    }
    __syncthreads();

    if (lane == 0) {
        float loss = 0.0f;
        for (int i = 0; i < Bn; ++i) {
            const float mi_bd = fmaxf(msum_bd[i], 1.0f);
            const float mi_bg = fmaxf(msum_bg[i], 1.0f);
            for (int j = 0; j < Bn; ++j) {
                const float sbb = G[j * 16 + j];
                const float sgg = G[(j + 4) * 16 + (j + 4)];
                const float sbg = G[j * 16 + (j + 4)];
                // x = fs_bd[j]/mi_bd ; y = fs_bg[j]/mi_bg
                const float dxx = sbb / (mi_bd * mi_bd);
                const float nx  = sqrtf(dxx);
                const float pos = dxx / fmaxf(nx * nx, EPSF);
                const float dxy = sbg / (mi_bd * mi_bg);
                const float ny  = sqrtf(sgg) / mi_bg;
                const float neg = dxy / fmaxf(nx * ny, EPSF);
                loss += -logf(expf(pos / TEMP)) - logf(1.0f - expf(neg / TEMP));
            }
        }
        out[0] = loss / (float)(Bn * Bn);
    }
}

// ---------------------------------------------------------------------------
extern "C" void kernel_launch(void* const* d_in, const int* in_sizes, int n_in,
                              void* d_out, int out_size, void* d_ws, size_t ws_size,
                              hipStream_t stream) {
    (void)in_sizes; (void)n_in; (void)out_size; (void)ws_size;
    const float* feat = (const float*)d_in[0];   // (B,C,H,W) f32
    const int*   tgt  = (const int*)d_in[1];     // (B,1,H,W) i32
    float* out = (float*)d_out;

    float* ws    = (float*)d_ws;
    float* bd    = ws;                                   // B*HW floats
    float* bg    = ws + (size_t)Bn * HWn;                // B*HW floats
    float* fs_bd = ws + 2 * (size_t)Bn * HWn;            // B*C
    float* fs_bg = fs_bd + Bn * Cn;                      // B*C
    float* ms_bd = fs_bg + Bn * Cn;                      // B
    float* ms_bg = ms_bd + Bn;                           // B
    const int n_acc = 2 * Bn * Cn + 2 * Bn;              // 520 floats

    bl_zero_kernel<<<(n_acc + 255) / 256, 256, 0, stream>>>(fs_bd, n_acc);
    bl_mask_kernel<<<(Bn * HWn) / 256, 256, 0, stream>>>(tgt, bd, bg, ms_bd, ms_bg);
    bl_reduce_kernel<<<dim3(Bn * Cn, 64), 256, 0, stream>>>(feat, bd, bg, fs_bd, fs_bg);
    bl_finalize_kernel<<<1, 32, 0, stream>>>(fs_bd, fs_bg, ms_bd, ms_bg, out);
}